// QuantumKernel_65481071395954
// MI455X (gfx1250) — compile-verified
//
#include <hip/hip_runtime.h>
#include <math.h>

// Reference collapses algebraically: output is the 16-byte row {c^2, 0, 0, 0}
// replicated B times, where c = prod over 24 theta entries of cos(theta/2).
// This is a pure 16 MB broadcast-fill: HBM-write-bandwidth bound (~0.7 us at
// 23.3 TB/s). No matrix math exists, so no WMMA; the CDNA5-relevant paths are
// wave32 sizing, 128-bit coalesced stores, and the async LDS->global store
// engine tracked by ASYNCcnt.

typedef float v4f __attribute__((ext_vector_type(4)));
typedef int   v4i __attribute__((vector_size(16)));   // matches builtin's pointee

#define RPT   8     // rows (16B each) per thread
#define BLOCK 256   // 8 wave32s per workgroup

// Prefer the gfx1250 async LDS->global store path if this toolchain exposes it.
#if __has_builtin(__builtin_amdgcn_global_store_async_from_lds_b128) && \
    __has_builtin(__builtin_amdgcn_s_wait_asynccnt)
#define USE_ASYNC_LDS_STORE 1
#else
#define USE_ASYNC_LDS_STORE 0
#endif

#if USE_ASYNC_LDS_STORE
// Generic->AS casts via integer round-trip (always accepted by clang).
// Low 32 bits of a generic LDS address are the LDS byte offset (ISA 10.2
// aperture mapping), so the truncating cast to the 32-bit AS(3) pointer is
// exactly the LDS offset the async store wants. Global flat addresses are
// identity-mapped to AS(1).
__device__ static inline __attribute__((address_space(1))) v4i*
as_global_v4(const void* p) {
  return (__attribute__((address_space(1))) v4i*)(unsigned long long)p;
}
__device__ static inline __attribute__((address_space(3))) v4i*
as_lds_v4(const void* p) {
  return (__attribute__((address_space(3))) v4i*)(unsigned int)(unsigned long long)p;
}
#endif

__global__ __launch_bounds__(BLOCK) void qprob_fill(
    const float* __restrict__ theta, int ntheta,
    float* __restrict__ out, long long nrows)
{
  // One 16-byte pattern per workgroup, parked in LDS.
  __shared__ __align__(16) float s_pat[4];

  if (threadIdx.x == 0) {
    float c = 1.0f;
    for (int i = 0; i < ntheta; ++i) c *= cosf(0.5f * theta[i]);
    s_pat[0] = c * c;   // probs column 0
    s_pat[1] = 0.0f;    // columns 1..3 are exactly zero
    s_pat[2] = 0.0f;
    s_pat[3] = 0.0f;
  }
  __syncthreads();

  const long long stride = (long long)gridDim.x * BLOCK;
  const long long gid    = (long long)blockIdx.x * BLOCK + threadIdx.x;

#if USE_ASYNC_LDS_STORE
  // Async DMA of the LDS pattern to global: per-lane b128, waves cover 512B
  // contiguous per instruction; no VGPR data movement; tracked by ASYNCcnt.
  #pragma unroll
  for (int i = 0; i < RPT; ++i) {
    long long row = gid + (long long)i * stride;
    if (row < nrows) {
      __builtin_amdgcn_global_store_async_from_lds_b128(
          as_global_v4(out + row * 4), as_lds_v4((const void*)s_pat),
          /*offset=*/0, /*cpol=*/0);
    }
  }
  __builtin_amdgcn_s_wait_asynccnt(0);
#else
  // Fallback: 128-bit nontemporal stores straight from VGPRs.
  v4f val;
  val.x = s_pat[0];
  val.y = 0.0f;
  val.z = 0.0f;
  val.w = 0.0f;
  #pragma unroll
  for (int i = 0; i < RPT; ++i) {
    long long row = gid + (long long)i * stride;
    if (row < nrows) {
      __builtin_nontemporal_store(val, (v4f*)(out + row * 4));
    }
  }
#endif
}

extern "C" void kernel_launch(void* const* d_in, const int* in_sizes, int n_in,
                              void* d_out, int out_size, void* d_ws, size_t ws_size,
                              hipStream_t stream) {
  // d_in[0] = patch (B,4) float32  -- provably unused by the reference math
  // d_in[1] = theta (3,4,2) float32 (24 elements)
  const float* theta  = (const float*)d_in[1];
  const int    ntheta = in_sizes[1];          // 24
  float*       out    = (float*)d_out;
  const long long nrows = (long long)in_sizes[0] / 4;   // B = 1,048,576

  long long nthreads = (nrows + RPT - 1) / RPT;
  int grid = (int)((nthreads + BLOCK - 1) / BLOCK);
  if (grid < 1) grid = 1;

  qprob_fill<<<grid, BLOCK, 0, stream>>>(theta, ntheta, out, nrows);
}